// DynamicConv1x1_47485158425386
// MI455X (gfx1250) — compile-verified
//
#include <hip/hip_runtime.h>
#include <stdint.h>

typedef float v2f __attribute__((ext_vector_type(2)));
typedef float v8f __attribute__((ext_vector_type(8)));

#define CIN   128
#define COUT  128
#define KEXP  4
#define HW    12544          // 112*112
#define NTILES_PER_BLOCK 16  // sixteen 16-wide N tiles per block
#define NCOLS (NTILES_PER_BLOCK * 16)   // 256 columns per block
#define CHUNKS (HW / NCOLS)             // 49

// ---------------- Kernel 1: global average pool --------------------------
// grid = B*Cin blocks, 256 threads. pooled[b*Cin+c] = mean(x[b,c,:,:])
__global__ __launch_bounds__(256) void pool_kernel(const float* __restrict__ x,
                                                   float* __restrict__ pooled) {
  __shared__ float red[256];
  const float* p = x + (size_t)blockIdx.x * HW;
  float s = 0.f;
  for (int i = threadIdx.x; i < HW; i += 256) s += p[i];
  red[threadIdx.x] = s;
  __syncthreads();
  for (int off = 128; off > 0; off >>= 1) {
    if ((int)threadIdx.x < off) red[threadIdx.x] += red[threadIdx.x + off];
    __syncthreads();
  }
  if (threadIdx.x == 0) pooled[blockIdx.x] = red[0] * (1.0f / (float)HW);
}

// ---------------- Kernel 2: gate softmax + weight combine ----------------
// grid = B blocks, 128 threads. wb[b] = sum_k softmax_k * expert_w[k]
__global__ __launch_bounds__(128) void gate_combine_kernel(
    const float* __restrict__ pooled, const float* __restrict__ gate_w,
    const float* __restrict__ gate_b, const float* __restrict__ expert_w,
    float* __restrict__ wb) {
  __shared__ float sp[KEXP][CIN];
  __shared__ float gsh[KEXP];
  const int b = blockIdx.x;
  const int t = threadIdx.x;

  const float pv = pooled[b * CIN + t];
#pragma unroll
  for (int k = 0; k < KEXP; ++k) sp[k][t] = pv * gate_w[k * CIN + t];
  __syncthreads();
  for (int off = 64; off > 0; off >>= 1) {
    if (t < off) {
#pragma unroll
      for (int k = 0; k < KEXP; ++k) sp[k][t] += sp[k][t + off];
    }
    __syncthreads();
  }
  if (t == 0) {
    float lg[KEXP], m = -3.0e38f;
#pragma unroll
    for (int k = 0; k < KEXP; ++k) { lg[k] = sp[k][0] + gate_b[k]; m = fmaxf(m, lg[k]); }
    float den = 0.f;
#pragma unroll
    for (int k = 0; k < KEXP; ++k) { lg[k] = __expf(lg[k] - m); den += lg[k]; }
    float inv = 1.f / den;
#pragma unroll
    for (int k = 0; k < KEXP; ++k) gsh[k] = lg[k] * inv;
  }
  __syncthreads();
  const float g0 = gsh[0], g1 = gsh[1], g2 = gsh[2], g3 = gsh[3];

  const float* w0 = expert_w + 0 * COUT * CIN + t * CIN;
  const float* w1 = expert_w + 1 * COUT * CIN + t * CIN;
  const float* w2 = expert_w + 2 * COUT * CIN + t * CIN;
  const float* w3 = expert_w + 3 * COUT * CIN + t * CIN;
  float* orow = wb + ((size_t)b * COUT + t) * CIN;
#pragma unroll 4
  for (int i = 0; i < CIN; i += 4) {
    float4 a = *(const float4*)(w0 + i);
    float4 bq = *(const float4*)(w1 + i);
    float4 c = *(const float4*)(w2 + i);
    float4 d = *(const float4*)(w3 + i);
    float4 r;
    r.x = g0 * a.x + g1 * bq.x + g2 * c.x + g3 * d.x;
    r.y = g0 * a.y + g1 * bq.y + g2 * c.y + g3 * d.y;
    r.z = g0 * a.z + g1 * bq.z + g2 * c.z + g3 * d.z;
    r.w = g0 * a.w + g1 * bq.w + g2 * c.w + g3 * d.w;
    *(float4*)(orow + i) = r;
  }
}

// ---------------- Kernel 3: batched GEMM via f32 WMMA --------------------
// Per batch b: out[b] (128 x 12544) = wb[b] (128x128) * x[b] (128x12544).
// Block = 8 waves; wave w owns M-tile w. A slab lives in VGPRs for the block
// lifetime; X tiles are double-buffered in LDS, filled by the CDNA5 async
// global->LDS DMA path (ASYNCcnt) so staging never touches VGPRs.
__global__ __launch_bounds__(256) void moe_gemm_kernel(
    const float* __restrict__ x, const float* __restrict__ wb,
    float* __restrict__ out) {
  __shared__ __align__(16) float xs[2][CIN * 16];  // 2 x 8 KB

  const int b     = blockIdx.x / CHUNKS;
  const int chunk = blockIdx.x % CHUNKS;
  const int tid   = threadIdx.x;
  const int wave  = tid >> 5;
  const int lane  = tid & 31;
  const int l16   = lane & 15;
  const int hi    = lane >> 4;          // lane half: selects K pair / M offset

  const float* xb = x   + (size_t)b * CIN  * HW;
  float*       ob = out + (size_t)b * COUT * HW;
  const float* Wb = wb  + (size_t)b * COUT * CIN;

  // ---- Load A fragments once: W rows for this wave's M-tile.
  // ISA 16x4 f32 A layout: lanes 0-15 -> (K=0,K=1), lanes 16-31 -> (K=2,K=3).
  const int arow = wave * 16 + l16;
  v2f afrag[32];
#pragma unroll
  for (int s = 0; s < 32; ++s) {
    afrag[s] = *(const v2f*)(Wb + (size_t)arow * CIN + 4 * s + 2 * hi);
  }

  const int n_base = chunk * NCOLS;

  // This thread's two 16-byte staging slots: vec index v -> row v>>2, colgrp v&3
  const int v0 = tid, v1 = tid + 256;
  const int so0 = (v0 >> 2) * 16 + (v0 & 3) * 4;   // float index within a buffer
  const int so1 = (v1 >> 2) * 16 + (v1 & 3) * 4;
  const size_t go0 = (size_t)(v0 >> 2) * HW + (v0 & 3) * 4;
  const size_t go1 = (size_t)(v1 >> 2) * HW + (v1 & 3) * 4;

  // Async DMA one 128x16 f32 tile (8 KB) into LDS buffer `buf`.
  auto async_stage = [&](int buf, int tile) {
    const float* src = xb + n_base + tile * 16;
    uint64_t g0 = (uint64_t)(uintptr_t)(src + go0);
    uint64_t g1 = (uint64_t)(uintptr_t)(src + go1);
    // Flat LDS address: addr[31:0] is the LDS byte offset (ISA 10.2).
    uint32_t l0 = (uint32_t)(uintptr_t)(&xs[buf][so0]);
    uint32_t l1 = (uint32_t)(uintptr_t)(&xs[buf][so1]);
    asm volatile("global_load_async_to_lds_b128 %0, %1, off\n\t"
                 "global_load_async_to_lds_b128 %2, %3, off"
                 :: "v"(l0), "v"(g0), "v"(l1), "v"(g1)
                 : "memory");
  };

  // Prologue: fill buffer 0 with tile 0.
  async_stage(0, 0);
  asm volatile("s_wait_asynccnt 0" ::: "memory");
  __syncthreads();

  for (int t = 0; t < NTILES_PER_BLOCK; ++t) {
    const int cur = t & 1;
    const int n0  = n_base + t * 16;
    const bool have_next = (t + 1) < NTILES_PER_BLOCK;

    // Kick off next tile's DMA before the WMMA chain (hidden by compute).
    if (have_next) {
      async_stage(1 - cur, t + 1);
      if (t + 2 < NTILES_PER_BLOCK) {
        __builtin_prefetch(xb + go0 + n0 + 32, 0, 1);   // global_prefetch_b8
      }
    }

    // ---- K=128 accumulation: 32x V_WMMA_F32_16X16X4_F32, two independent
    // accumulator chains for XDL pipelining.
    v8f acc0 = {}, acc1 = {};
#pragma unroll
    for (int s = 0; s < 32; s += 2) {
      const int k0 = 4 * s + 2 * hi;        // B 4x16 layout: VGPR0=K0/K2, VGPR1=K1/K3
      v2f bf0; bf0.x = xs[cur][k0 * 16 + l16];
               bf0.y = xs[cur][(k0 + 1) * 16 + l16];
      acc0 = __builtin_amdgcn_wmma_f32_16x16x4_f32(
          false, afrag[s], false, bf0, (short)0, acc0, false, false);

      const int k1 = 4 * (s + 1) + 2 * hi;
      v2f bf1; bf1.x = xs[cur][k1 * 16 + l16];
               bf1.y = xs[cur][(k1 + 1) * 16 + l16];
      acc1 = __builtin_amdgcn_wmma_f32_16x16x4_f32(
          false, afrag[s + 1], false, bf1, (short)0, acc1, false, false);
    }

    // ---- Write 16x16 D tile: VGPR j -> M=j (lanes 0-15) / M=j+8 (lanes 16-31)
#pragma unroll
    for (int j = 0; j < 8; ++j) {
      const int orow = wave * 16 + j + 8 * hi;
      ob[(size_t)orow * HW + n0 + l16] = acc0[j] + acc1[j];
    }

    if (have_next) {
      asm volatile("s_wait_asynccnt 0" ::: "memory");  // own DMA done
    }
    __syncthreads();                                    // all waves' DMA done
  }
}

// ---------------- Launch ----------------
extern "C" void kernel_launch(void* const* d_in, const int* in_sizes, int n_in,
                              void* d_out, int out_size, void* d_ws, size_t ws_size,
                              hipStream_t stream) {
  const float* x        = (const float*)d_in[0];  // [B,128,112,112]
  const float* expert_w = (const float*)d_in[1];  // [4,128,128]
  const float* gate_w   = (const float*)d_in[2];  // [4,128]
  const float* gate_b   = (const float*)d_in[3];  // [4]
  float* out = (float*)d_out;

  const int B = in_sizes[0] / (CIN * HW);         // 32

  float* pooled = (float*)d_ws;                   // B*Cin floats
  float* wbuf   = pooled + (size_t)B * CIN;       // B*Cout*Cin floats (~2 MB)

  pool_kernel<<<B * CIN, 256, 0, stream>>>(x, pooled);
  gate_combine_kernel<<<B, 128, 0, stream>>>(pooled, gate_w, gate_b, expert_w, wbuf);
  moe_gemm_kernel<<<B * CHUNKS, 256, 0, stream>>>(x, wbuf, out);
}